// BatchTopKTiedSAE_86431921864930
// MI455X (gfx1250) — compile-verified
//
#include <hip/hip_runtime.h>

// BatchTopK tied SAE forward for MI455X (gfx1250, wave32).
//   enc:  h = relu(x @ W + b_enc)          [4096 x 16384]  (WMMA f32 GEMM)
//   topk: per-row radix-select top-64 of (h + tiebreaker), keep h values
//   dec:  recon = f @ W^T + b_dec          [4096 x 8192]   (WMMA f32 GEMM)
//
// Both GEMMs are HBM-bound (W = 512MB fp32 per GEMM at 23.3 TB/s). Grid is
// ordered m-fastest so co-resident blocks share one W column-slab (streamed
// from HBM once; x/f re-reads hit the 192MB L2). Tiles are DMA'd with
// GLOBAL_LOAD_ASYNC_TO_LDS_B128 into double-buffered LDS, overlapped with
// V_WMMA_F32_16X16X4_F32 compute, synced via s_wait_asynccnt.

#define D_IN   8192
#define D_HID  16384
#define NROWS  4096
#define TOPK   64

#define BM 64
#define BN 128
#define BK 16

typedef __attribute__((ext_vector_type(2))) float v2f;
typedef __attribute__((ext_vector_type(8))) float v8f;

// ---------------------------------------------------------------------------
// CDNA5 async global->LDS copy (ASYNCcnt-tracked), 16B per lane.
// Low 32 bits of a flat shared pointer are the LDS byte address on gfx1250.
// ---------------------------------------------------------------------------
__device__ __forceinline__ void async_copy_b128(void* lds_dst, const void* gsrc) {
  unsigned dst = (unsigned)(size_t)lds_dst;
  asm volatile("global_load_async_to_lds_b128 %0, %1, off"
               :: "v"(dst), "v"(gsrc)
               : "memory");
}
__device__ __forceinline__ void async_wait0() {
  asm volatile("s_wait_asynccnt 0" ::: "memory");
}

// ---------------------------------------------------------------------------
// WMMA inner slab: BK=16 -> 4 k-steps x 4 n-subtiles = 16 v_wmma per call.
// A-frag (16x4 f32): lanes 0-15 hold K={0,1}, lanes 16-31 K={2,3}.
// B-frag (4x16 f32): vgpr0 rows K=0/2 (per lane half), vgpr1 rows K=1/3.
// ---------------------------------------------------------------------------
__device__ __forceinline__ void wmma_slab(const float (*sA)[20],
                                          const float (*sB)[BN + 4],
                                          int wm, int wn, int lh, int ll,
                                          v8f acc[4]) {
#pragma unroll
  for (int kk = 0; kk < 4; ++kk) {
    const int ka = kk * 4 + lh * 2;
    v2f a;
    a.x = sA[wm * 16 + ll][ka];
    a.y = sA[wm * 16 + ll][ka + 1];
#pragma unroll
    for (int t = 0; t < 4; ++t) {
      const int nb = wn * 64 + t * 16 + ll;
      v2f b;
      b.x = sB[ka][nb];
      b.y = sB[ka + 1][nb];
      acc[t] = __builtin_amdgcn_wmma_f32_16x16x4_f32(
          false, a, false, b, (short)0, acc[t], false, false);
    }
  }
}

// Decoder variant: B tile stored un-transposed as sBt[n][k] (stride 20 floats
// keeps the 16-lane bank pattern {20n mod 64} conflict-free); transpose
// happens in the fragment read.
__device__ __forceinline__ void wmma_slab_t(const float (*sA)[20],
                                            const float (*sBt)[20],
                                            int wm, int wn, int lh, int ll,
                                            v8f acc[4]) {
#pragma unroll
  for (int kk = 0; kk < 4; ++kk) {
    const int ka = kk * 4 + lh * 2;
    v2f a;
    a.x = sA[wm * 16 + ll][ka];
    a.y = sA[wm * 16 + ll][ka + 1];
#pragma unroll
    for (int t = 0; t < 4; ++t) {
      const int nb = wn * 64 + t * 16 + ll;
      v2f b;
      b.x = sBt[nb][ka];
      b.y = sBt[nb][ka + 1];
      acc[t] = __builtin_amdgcn_wmma_f32_16x16x4_f32(
          false, a, false, b, (short)0, acc[t], false, false);
    }
  }
}

// ---------------------------------------------------------------------------
// Encoder GEMM: h[m,n] = relu( sum_k x[m,k] * W[k,n] + b_enc[n] )
// ---------------------------------------------------------------------------
__global__ __launch_bounds__(256) void enc_gemm(const float* __restrict__ x,
                                                const float* __restrict__ W,
                                                const float* __restrict__ b_enc,
                                                float* __restrict__ h) {
  __shared__ float sA[2][BM][20];        // 64x16 x-tile, double buffered
  __shared__ float sB[2][BK][BN + 4];    // 16x128 W-tile, double buffered

  const int tid  = threadIdx.x;
  const int lane = tid & 31;
  const int wave = tid >> 5;
  const int wm = wave & 3;
  const int wn = wave >> 2;
  const int lh = lane >> 4;
  const int ll = lane & 15;
  const int m0 = blockIdx.x * BM;        // m fastest: W slab shared by
  const int n0 = blockIdx.y * BN;        // co-resident blocks

  v8f acc[4] = {};

  const int arow = tid >> 2;             // 0..63
  const int acol = (tid & 3) << 2;       // 0,4,8,12
  const int brow = tid >> 5;             // 0..7 (+8 second copy)
  const int bcol = (tid & 31) << 2;      // 0..124

  const float* xg = x + (size_t)(m0 + arow) * D_IN + acol;
  const float* wg = W + (size_t)brow * D_HID + (n0 + bcol);

  auto load_tiles = [&](int b, int k0) {
    async_copy_b128(&sA[b][arow][acol], xg + k0);
    async_copy_b128(&sB[b][brow][bcol], wg + (size_t)k0 * D_HID);
    async_copy_b128(&sB[b][brow + 8][bcol], wg + (size_t)(k0 + 8) * D_HID);
  };

  load_tiles(0, 0);
  async_wait0();
  __syncthreads();

  const int NT = D_IN / BK;              // 512 tiles
  for (int it = 0; it < NT; ++it) {
    const int cur = it & 1;
    if (it + 1 < NT) load_tiles(cur ^ 1, (it + 1) * BK);
    wmma_slab(sA[cur], sB[cur], wm, wn, lh, ll, acc);
    async_wait0();
    __syncthreads();
  }

  float* hout = h + (size_t)m0 * D_HID + n0;
#pragma unroll
  for (int t = 0; t < 4; ++t) {
    const int col   = wn * 64 + t * 16 + ll;
    const float bias = b_enc[n0 + col];
#pragma unroll
    for (int r = 0; r < 8; ++r) {
      const int row = wm * 16 + r + lh * 8;
      float v = acc[t][r] + bias;
      hout[(size_t)row * D_HID + col] = v > 0.0f ? v : 0.0f;
    }
  }
}

// ---------------------------------------------------------------------------
// Decoder GEMM: recon[m,n] = sum_j f[m,j] * W[n,j] + b_dec[n]
// W tile async-copied row-major (j-contiguous) into sBt[n][j].
// ---------------------------------------------------------------------------
__global__ __launch_bounds__(256) void dec_gemm(const float* __restrict__ f,
                                                const float* __restrict__ W,
                                                const float* __restrict__ b_dec,
                                                float* __restrict__ recon) {
  __shared__ float sA[2][BM][20];        // 64x16 f-tile
  __shared__ float sBt[2][BN][20];       // 128x16 W-tile (un-transposed)

  const int tid  = threadIdx.x;
  const int lane = tid & 31;
  const int wave = tid >> 5;
  const int wm = wave & 3;
  const int wn = wave >> 2;
  const int lh = lane >> 4;
  const int ll = lane & 15;
  const int m0 = blockIdx.x * BM;
  const int n0 = blockIdx.y * BN;        // over D_IN

  v8f acc[4] = {};

  const int arow = tid >> 2;             // 0..63
  const int acol = (tid & 3) << 2;       // 0,4,8,12
  const int nrow = tid >> 2;             // 0..63 (+64 second copy)
  const int jcol = (tid & 3) << 2;       // 0,4,8,12

  const float* ag = f + (size_t)(m0 + arow) * D_HID + acol;
  const float* wg0 = W + (size_t)(n0 + nrow) * D_HID + jcol;
  const float* wg1 = W + (size_t)(n0 + nrow + 64) * D_HID + jcol;

  auto load_tiles = [&](int b, int k0) {
    async_copy_b128(&sA[b][arow][acol], ag + k0);
    async_copy_b128(&sBt[b][nrow][jcol], wg0 + k0);
    async_copy_b128(&sBt[b][nrow + 64][jcol], wg1 + k0);
  };

  load_tiles(0, 0);
  async_wait0();
  __syncthreads();

  const int NT = D_HID / BK;             // 1024 tiles
  for (int it = 0; it < NT; ++it) {
    const int cur = it & 1;
    if (it + 1 < NT) load_tiles(cur ^ 1, (it + 1) * BK);
    wmma_slab_t(sA[cur], sBt[cur], wm, wn, lh, ll, acc);
    async_wait0();
    __syncthreads();
  }

  float* rout = recon + (size_t)m0 * D_IN + n0;
#pragma unroll
  for (int t = 0; t < 4; ++t) {
    const int col   = wn * 64 + t * 16 + ll;
    const float bias = b_dec[n0 + col];
#pragma unroll
    for (int r = 0; r < 8; ++r) {
      const int row = wm * 16 + r + lh * 8;
      rout[(size_t)row * D_IN + col] = acc[t][r] + bias;
    }
  }
}

// ---------------------------------------------------------------------------
// Top-K mask: per row, MSB-first radix select threshold on order-preserving
// uint keys of (h + tiebreaker); keep original h where key >= threshold.
// ---------------------------------------------------------------------------
__device__ __forceinline__ unsigned fkey(float x) {
  unsigned u = __float_as_uint(x);
  return (u & 0x80000000u) ? ~u : (u | 0x80000000u);
}

__global__ __launch_bounds__(256) void topk_mask(float* __restrict__ f,
                                                 const float* __restrict__ tb) {
  const int row = blockIdx.x;
  float* hrow = f + (size_t)row * D_HID;
  __shared__ unsigned hist[256];
  __shared__ unsigned s_sel, s_want;
  const int tid = threadIdx.x;

  unsigned prefix = 0;
  unsigned want = TOPK;
  for (int shift = 24; shift >= 0; shift -= 8) {
    hist[tid] = 0;
    __syncthreads();
    const unsigned mask = (shift == 24) ? 0u : (0xFFFFFFFFu << (shift + 8));
    for (int i = tid; i < D_HID; i += 256) {
      const unsigned key = fkey(hrow[i] + tb[i]);
      if ((key & mask) == prefix)
        atomicAdd(&hist[(key >> shift) & 255u], 1u);
    }
    __syncthreads();
    if (tid == 0) {
      unsigned cum = 0;
      int d = 255;
      for (; d > 0; --d) {
        if (cum + hist[d] >= want) break;
        cum += hist[d];
      }
      s_sel  = (unsigned)d;
      s_want = want - cum;
    }
    __syncthreads();
    prefix |= s_sel << shift;
    want = s_want;
    __syncthreads();
  }

  const unsigned T = prefix;             // key of 64th-largest element
  for (int i = tid; i < D_HID; i += 256) {
    const float hv = hrow[i];
    const unsigned key = fkey(hv + tb[i]);
    hrow[i] = (key >= T) ? hv : 0.0f;
  }
}

// ---------------------------------------------------------------------------
extern "C" void kernel_launch(void* const* d_in, const int* in_sizes, int n_in,
                              void* d_out, int out_size, void* d_ws,
                              size_t ws_size, hipStream_t stream) {
  const float* x  = (const float*)d_in[0];
  const float* W  = (const float*)d_in[1];
  const float* be = (const float*)d_in[2];
  const float* bd = (const float*)d_in[3];
  const float* tb = (const float*)d_in[4];

  float* recon = (float*)d_out;                    // [NROWS, D_IN]
  float* f     = recon + (size_t)NROWS * D_IN;     // [NROWS, D_HID]

  enc_gemm<<<dim3(NROWS / BM, D_HID / BN), dim3(256), 0, stream>>>(x, W, be, f);
  topk_mask<<<dim3(NROWS), dim3(256), 0, stream>>>(f, tb);
  dec_gemm<<<dim3(NROWS / BM, D_IN / BN), dim3(256), 0, stream>>>(f, W, bd, recon);
}